// Repro_89850715832655
// MI455X (gfx1250) — compile-verified
//
#include <hip/hip_runtime.h>
#include <hip/hip_bf16.h>
#include <math.h>

#define B_  16
#define D_  256
#define H_  8
#define HD_ 32
#define W_  512
#define EPS_ 1.1920928955078125e-07f

typedef __attribute__((ext_vector_type(2))) float v2f;
typedef __attribute__((ext_vector_type(8))) float v8f;

// ---------------------------------------------------------------------------
// Kernel A: x = emb[token] + pos_emb[W-1];  h = rmsnorm(x) * rms_w
// grid = (16), block = (256)
// ---------------------------------------------------------------------------
__global__ void embed_rms_kernel(const float* __restrict__ emb,
                                 const int*   __restrict__ tokens,
                                 const float* __restrict__ pos_emb,
                                 const float* __restrict__ rms_w,
                                 float* __restrict__ xout,
                                 float* __restrict__ hout) {
  __shared__ float red[D_];
  const int b = blockIdx.x;
  const int d = threadIdx.x;
  const long long row = (long long)tokens[b] * D_;
  const float x = emb[row + d] + pos_emb[(long long)(W_ - 1) * D_ + d];
  red[d] = x * x;
  __syncthreads();
  for (int s = D_ / 2; s > 0; s >>= 1) {
    if (d < s) red[d] += red[d + s];
    __syncthreads();
  }
  const float inv = rsqrtf(red[0] / (float)D_ + EPS_);
  xout[b * D_ + d] = x;
  hout[b * D_ + d] = x * inv * rms_w[d];
}

// ---------------------------------------------------------------------------
// Kernel B: q/k/v projections via V_WMMA_F32_16X16X4_F32 (fp32-exact WMMA)
// One wave (32 threads) per 16x16 output tile.
//   tiles 0..15  -> q  (16x256), tiles 16..17 -> k_new (16x32),
//   tiles 18..33 -> v  (16x256)
// A (16x4 f32): lane m = lane&15, VGPR0/1 = K = kb + 2*(lane>=16) + {0,1}
// B (4x16 f32): lane n = lane&15, VGPR0/1 = K = kb + 2*(lane>=16) + {0,1}
//               B[k][n] = w[n][k] (w row-major (N,K)) -> contiguous float2
// C/D: VGPR r -> row r + 8*(lane>=16), col = n0 + (lane&15)
// ---------------------------------------------------------------------------
__global__ void qkv_wmma_kernel(const float* __restrict__ h,
                                const float* __restrict__ wq, const float* __restrict__ bq,
                                const float* __restrict__ wk, const float* __restrict__ bk,
                                const float* __restrict__ wv, const float* __restrict__ bv,
                                float* __restrict__ q,
                                float* __restrict__ kn,
                                float* __restrict__ vn) {
  const int tile = blockIdx.x;
  const float* wmat;
  const float* bias;
  float* out;
  int n0, ldo;
  if (tile < 16)      { wmat = wq; bias = bq; out = q;  n0 = tile * 16;        ldo = D_;  }
  else if (tile < 18) { wmat = wk; bias = bk; out = kn; n0 = (tile - 16) * 16; ldo = HD_; }
  else                { wmat = wv; bias = bv; out = vn; n0 = (tile - 18) * 16; ldo = D_;  }

  const int lane = threadIdx.x;          // 0..31, one wave
  const int m    = lane & 15;            // row (A) / col (B)
  const int koff = (lane >> 4) << 1;     // 0 or 2: K half

  v8f c = {};
  const float* arow = h + m * D_;
  const float* brow = wmat + (n0 + m) * D_;
  for (int kb = 0; kb < D_; kb += 4) {
    v2f a = *(const v2f*)(arow + kb + koff);
    v2f b = *(const v2f*)(brow + kb + koff);
    c = __builtin_amdgcn_wmma_f32_16x16x4_f32(
        /*neg_a=*/false, a, /*neg_b=*/false, b,
        /*c_mod=*/(short)0, c, /*reuse_a=*/false, /*reuse_b=*/false);
  }

  const float bb = bias[n0 + m];
  const int rbase = (lane >> 4) * 8;
#pragma unroll
  for (int r = 0; r < 8; ++r) {
    out[(rbase + r) * ldo + n0 + m] = c[r] + bb;
  }
}

// ---------------------------------------------------------------------------
// Kernel C: RoPE(q @ pos 1, k @ pos w) + MQA scores + softmax + attn.V + x
// grid = (16, 8)  block = (512): thread t = key position w for scores,
// then (chunk = t>>5, hd = t&31) for the attn.V reduction.
// v_all raw reshape: head h reads v rows h*64 + (w'>>3), d = (w'&7)*32 + hd.
// ---------------------------------------------------------------------------
__global__ void attn_kernel(const float* __restrict__ q,
                            const float* __restrict__ kn,
                            const float* __restrict__ vn,
                            const float* __restrict__ kc,   // (B, W-1, HD)
                            const float* __restrict__ vc,   // (B, W-1, D)
                            const float* __restrict__ x,
                            const float* __restrict__ cosT, // (W, HD)
                            const float* __restrict__ sinT,
                            const float* __restrict__ mask, // (1,1,W,W); row 0 used
                            float* __restrict__ out) {
  const int b  = blockIdx.x;
  const int hI = blockIdx.y;
  const int t  = threadIdx.x;            // 0..511

  __shared__ float qr[HD_];
  __shared__ float attn[W_];
  __shared__ float red[W_];
  __shared__ float part[16][HD_];

  // RoPE on q: reference uses cos[1:2]/sin[1:2] (position index 1)
  if (t < 16) {
    const float te = q[b * D_ + hI * HD_ + 2 * t];
    const float to = q[b * D_ + hI * HD_ + 2 * t + 1];
    const float ce = cosT[1 * HD_ + 2 * t];
    const float se = sinT[1 * HD_ + 2 * t];
    qr[2 * t]     = te * ce - to * se;
    qr[2 * t + 1] = te * se + to * ce;
  }
  __syncthreads();

  // Score for key position w = t (RoPE applied on the fly)
  const int w = t;
  const float* krow = (w < W_ - 1) ? (kc + ((long long)b * (W_ - 1) + w) * HD_)
                                   : (kn + b * HD_);
  float s = 0.f;
#pragma unroll
  for (int i = 0; i < 16; ++i) {
    const float te = krow[2 * i];
    const float to = krow[2 * i + 1];
    const float ce = cosT[w * HD_ + 2 * i];
    const float se = sinT[w * HD_ + 2 * i];
    const float ke = te * ce - to * se;
    const float ko = te * se + to * ce;
    s += qr[2 * i] * ke + qr[2 * i + 1] * ko;
  }
  s *= 0.17677669529663687f;             // 1/sqrt(32)
  if (mask[w] == 0.f) s = -INFINITY;     // mask[:, :, 0:1] row

  // softmax: block max
  red[t] = s;
  __syncthreads();
  for (int st = W_ / 2; st > 0; st >>= 1) {
    if (t < st) red[t] = fmaxf(red[t], red[t + st]);
    __syncthreads();
  }
  const float mx = red[0];
  __syncthreads();
  const float e = expf(s - mx);
  red[t] = e;
  __syncthreads();
  for (int st = W_ / 2; st > 0; st >>= 1) {
    if (t < st) red[t] += red[t + st];
    __syncthreads();
  }
  attn[t] = e * (1.f / red[0]);
  __syncthreads();

  // attn . V with the raw-reshape v layout
  const int hd    = t & 31;
  const int chunk = t >> 5;              // 0..15
  float acc = 0.f;
#pragma unroll 8
  for (int j = 0; j < 32; ++j) {
    const int wp  = chunk * 32 + j;            // w' in 0..511
    const int wv2 = hI * 64 + (wp >> 3);       // v row
    const int dd  = (wp & 7) * HD_ + hd;       // v col
    const float* vrow = (wv2 < W_ - 1) ? (vc + ((long long)b * (W_ - 1) + wv2) * D_)
                                       : (vn + b * D_);
    acc += attn[wp] * vrow[dd];
  }
  part[chunk][hd] = acc;
  __syncthreads();

  if (t < HD_) {
    float o = 0.f;
#pragma unroll
    for (int cI = 0; cI < 16; ++cI) o += part[cI][t];
    out[b * D_ + hI * HD_ + t] = o + x[b * D_ + hI * HD_ + t];
  }
}

// ---------------------------------------------------------------------------
extern "C" void kernel_launch(void* const* d_in, const int* in_sizes, int n_in,
                              void* d_out, int out_size, void* d_ws, size_t ws_size,
                              hipStream_t stream) {
  const float* emb_table = (const float*)d_in[0];
  const int*   tokens    = (const int*)  d_in[1];
  const float* pos_emb   = (const float*)d_in[2];
  const float* rms_w     = (const float*)d_in[3];
  const float* wq        = (const float*)d_in[4];
  const float* bq        = (const float*)d_in[5];
  const float* wk        = (const float*)d_in[6];
  const float* bk        = (const float*)d_in[7];
  const float* wv        = (const float*)d_in[8];
  const float* bv        = (const float*)d_in[9];
  const float* k_cache   = (const float*)d_in[10];
  const float* v_cache   = (const float*)d_in[11];
  const float* cosT      = (const float*)d_in[12];
  const float* sinT      = (const float*)d_in[13];
  const float* mask      = (const float*)d_in[14];
  float* out = (float*)d_out;

  // workspace layout (floats)
  float* ws = (float*)d_ws;
  float* x_buf = ws;                 // 16*256
  float* h_buf = x_buf + B_ * D_;    // 16*256
  float* q_buf = h_buf + B_ * D_;    // 16*256
  float* k_buf = q_buf + B_ * D_;    // 16*32
  float* v_buf = k_buf + B_ * HD_;   // 16*256

  embed_rms_kernel<<<B_, D_, 0, stream>>>(emb_table, tokens, pos_emb, rms_w,
                                          x_buf, h_buf);

  qkv_wmma_kernel<<<34, 32, 0, stream>>>(h_buf, wq, bq, wk, bk, wv, bv,
                                         q_buf, k_buf, v_buf);

  attn_kernel<<<dim3(B_, H_), W_, 0, stream>>>(q_buf, k_buf, v_buf,
                                               k_cache, v_cache, x_buf,
                                               cosT, sinT, mask, out);
}